// WeightedHausdorffDistance_7997229105886
// MI455X (gfx1250) — compile-verified
//
#include <hip/hip_runtime.h>
#include <cstdint>

#define N_POSK 8192
#define BATCHK 4
#define N_GTK  512
#define EPSK   1e-6f

// ws layout (uint32/float slots):
//  u[0]: max(dis) bits   u[1]: min(prob) bits   u[2]: max(prob) bits
//  f[8..12): term1 numerator per batch
//  f[16..16+2048): S[b*512+g] accumulators
#define WS_MAXD  0
#define WS_PMIN  1
#define WS_PMAX  2
#define WS_T1    8
#define WS_S     16
#define WS_TOTAL (WS_S + BATCHK * N_GTK)

#define MAIN_GRID    256
#define ROWS_PER_BLK (N_POSK / MAIN_GRID)  // 32

// ---------------------------------------------------------------- init ----
__global__ void k_init(uint32_t* ws) {
  int i = blockIdx.x * blockDim.x + threadIdx.x;
  if (i == WS_PMIN)      ws[i] = 0x7F800000u;  // +inf bits
  else if (i < WS_TOTAL) ws[i] = 0u;           // max=0, S=0.0f, t1=0.0f
}

// --------------------------------------------------------- max(dis) ------
__global__ __launch_bounds__(256) void k_maxdis(const float4* __restrict__ dis,
                                                uint32_t* __restrict__ ws) {
  int tid = blockIdx.x * blockDim.x + threadIdx.x;
  int stride = gridDim.x * blockDim.x;
  const int total = (N_POSK * N_POSK) / 4;
  float m = 0.0f;  // values are uniform [0,1): non-negative
  for (int i = tid; i < total; i += stride) {
    float4 v = dis[i];
    m = fmaxf(m, fmaxf(fmaxf(v.x, v.y), fmaxf(v.z, v.w)));
  }
  for (int off = 16; off; off >>= 1) m = fmaxf(m, __shfl_xor(m, off, 32));
  __shared__ float wm[8];
  if ((threadIdx.x & 31) == 0) wm[threadIdx.x >> 5] = m;
  __syncthreads();
  if (threadIdx.x == 0) {
    float bm = wm[0];
    for (int w = 1; w < 8; ++w) bm = fmaxf(bm, wm[w]);
    atomicMax(&ws[WS_MAXD], __float_as_uint(bm));  // bit-monotone for x>=0
  }
}

// ----------------------------------------------------- prob min/max ------
__global__ __launch_bounds__(256) void k_probminmax(const float* __restrict__ prob,
                                                    uint32_t* __restrict__ ws) {
  int tid = blockIdx.x * blockDim.x + threadIdx.x;
  int stride = gridDim.x * blockDim.x;
  float mn = 3.4e38f, mx = 0.0f;
  for (int i = tid; i < BATCHK * N_POSK; i += stride) {
    float v = prob[i];
    mn = fminf(mn, v);
    mx = fmaxf(mx, v);
  }
  for (int off = 16; off; off >>= 1) {
    mn = fminf(mn, __shfl_xor(mn, off, 32));
    mx = fmaxf(mx, __shfl_xor(mx, off, 32));
  }
  if ((threadIdx.x & 31) == 0) {
    atomicMin(&ws[WS_PMIN], __float_as_uint(mn));
    atomicMax(&ws[WS_PMAX], __float_as_uint(mx));
  }
}

// --------------------------------------------------- async row stage -----
// One call per wave issues 8 x global_load_async_to_lds_b128 (ASYNCcnt += 8).
// 256 threads cover one full 32KB row: chunk c = t + 256*k, 16B each.
__device__ __forceinline__ void async_row_load(const float* __restrict__ dis,
                                               int row, uint32_t lds_base, int t) {
  const uint8_t* gbase = (const uint8_t*)(dis + (size_t)row * N_POSK);
#pragma unroll
  for (int k = 0; k < 8; ++k) {
    int c = t + 256 * k;
    uint32_t lds = lds_base + (uint32_t)c * 16u;
    uint64_t ga  = (uint64_t)(gbase + (size_t)c * 16u);
    asm volatile("global_load_async_to_lds_b128 %0, %1, off"
                 :: "v"(lds), "v"(ga) : "memory");
  }
}

// -------------------------------------------------------------- main -----
__global__ __launch_bounds__(256) void k_main(const float* __restrict__ dis,
                                              const float* __restrict__ prob_map,
                                              const int* __restrict__ gt,
                                              uint32_t* __restrict__ wsu) {
  extern __shared__ float smemf[];                 // dynamic LDS, base offset 0
  float* buf0 = smemf;                             // LDS byte offset 0
  float* buf1 = smemf + N_POSK;                    // LDS byte offset 32768
  float* wmin = smemf + 2 * N_POSK;                // LDS byte offset 65536 (8 floats)

  const int t  = threadIdx.x;
  const int b  = t >> 6;        // waves {2b, 2b+1} handle batch b
  const int gl = t & 63;        // this thread's base g-lane

  int cols[8];
#pragma unroll
  for (int k = 0; k < 8; ++k) cols[k] = gt[b * N_GTK + gl + 64 * k];

  const float maxd = __uint_as_float(wsu[WS_MAXD]);
  const float pmin = __uint_as_float(wsu[WS_PMIN]);
  const float pr   = __uint_as_float(wsu[WS_PMAX]) - pmin;

  float S[8];
#pragma unroll
  for (int k = 0; k < 8; ++k) S[k] = 0.0f;
  float t1 = 0.0f;  // used by threads 0..3 (term1 partial for batch t)

  const int row0 = blockIdx.x * ROWS_PER_BLK;
  async_row_load(dis, row0, 0u, t);  // prefetch first row into buf0
  int cur = 0;

  for (int i = 0; i < ROWS_PER_BLK; ++i) {
    const int row = row0 + i;
    __syncthreads();  // prior processing done: safe to overwrite next buf + wmin
    if (i + 1 < ROWS_PER_BLK) {
      async_row_load(dis, row + 1, cur ? 0u : 32768u, t);
      asm volatile("s_wait_asynccnt 8" ::: "memory");  // current row's 8 done (in-order)
    } else {
      asm volatile("s_wait_asynccnt 0" ::: "memory");
    }
    __syncthreads();  // current row visible to all waves

    const float* rb = cur ? buf1 : buf0;
    float p = fminf(fmaxf((prob_map[b * N_POSK + row] - pmin) / pr, 0.0f), 1.0f);
    float wb = (1.0f - p) * maxd;
    float dmin = 3.4e38f;
#pragma unroll
    for (int k = 0; k < 8; ++k) {
      float d = rb[cols[k]];                 // ds_load_b32 gather from LDS
      dmin = fminf(dmin, d);
      float w = fmaf(p, d, wb) + EPSK;       // (1-p)*maxd + p*d + eps
      S[k] += 1.0f / w;                      // (w+eps)^P_EXP, P_EXP = -1
    }
#pragma unroll
    for (int off = 16; off; off >>= 1) dmin = fminf(dmin, __shfl_xor(dmin, off, 32));
    if ((t & 31) == 0) wmin[t >> 5] = dmin;
    __syncthreads();
    if (t < BATCHK) {
      float mb = fminf(wmin[2 * t], wmin[2 * t + 1]);   // min_g over 512 for batch t
      float pb = fminf(fmaxf((prob_map[t * N_POSK + row] - pmin) / pr, 0.0f), 1.0f);
      t1 += pb * mb;
    }
    cur ^= 1;
  }

  float* wsf = (float*)wsu;
#pragma unroll
  for (int k = 0; k < 8; ++k)
    atomicAdd(&wsf[WS_S + b * N_GTK + gl + 64 * k], S[k]);
  if (t < BATCHK) atomicAdd(&wsf[WS_T1 + t], t1);
}

// ---------------------------------------------------------- finalize -----
__global__ __launch_bounds__(256) void k_final(const float* __restrict__ prob_map,
                                               const uint32_t* __restrict__ wsu,
                                               float* __restrict__ out) {
  const float* wsf = (const float*)wsu;
  const float pmin = __uint_as_float(wsu[WS_PMIN]);
  const float pr   = __uint_as_float(wsu[WS_PMAX]) - pmin;
  const int t = threadIdx.x;
  __shared__ float red[8];
  float nest[BATCHK];

  for (int bb = 0; bb < BATCHK; ++bb) {
    float s = 0.0f;
    for (int n = t; n < N_POSK; n += 256) {
      float p = fminf(fmaxf((prob_map[bb * N_POSK + n] - pmin) / pr, 0.0f), 1.0f);
      s += p;
    }
    for (int off = 16; off; off >>= 1) s += __shfl_xor(s, off, 32);
    if ((t & 31) == 0) red[t >> 5] = s;
    __syncthreads();
    if (t == 0) { float a = 0.0f; for (int w = 0; w < 8; ++w) a += red[w]; red[0] = a; }
    __syncthreads();
    nest[bb] = red[0];
    __syncthreads();
  }

  // term2 = mean_{b,g} ( N / S[b,g] )   since minn = (S/N)^(-1)
  float s2 = 0.0f;
  for (int i = t; i < BATCHK * N_GTK; i += 256)
    s2 += (float)N_POSK / wsf[WS_S + i];
  for (int off = 16; off; off >>= 1) s2 += __shfl_xor(s2, off, 32);
  if ((t & 31) == 0) red[t >> 5] = s2;
  __syncthreads();
  if (t == 0) {
    float a = 0.0f;
    for (int w = 0; w < 8; ++w) a += red[w];
    float term2 = a / (float)(BATCHK * N_GTK);
    float term1 = 0.0f;
    for (int bb = 0; bb < BATCHK; ++bb)
      term1 += wsf[WS_T1 + bb] / (nest[bb] + EPSK);
    term1 /= (float)BATCHK;
    out[0] = term1 + term2;
  }
}

// ------------------------------------------------------------ launch -----
extern "C" void kernel_launch(void* const* d_in, const int* in_sizes, int n_in,
                              void* d_out, int out_size, void* d_ws, size_t ws_size,
                              hipStream_t stream) {
  const float* prob = (const float*)d_in[0];   // (4, 8192) f32
  const int*   gt   = (const int*)d_in[1];     // (4, 512)  i32
  const float* dis  = (const float*)d_in[2];   // (8192, 8192) f32
  uint32_t* ws  = (uint32_t*)d_ws;
  float*    out = (float*)d_out;

  k_init<<<(WS_TOTAL + 255) / 256, 256, 0, stream>>>(ws);
  k_maxdis<<<2048, 256, 0, stream>>>((const float4*)dis, ws);
  k_probminmax<<<32, 256, 0, stream>>>(prob, ws);
  k_main<<<MAIN_GRID, 256, (2 * N_POSK + 8) * sizeof(float), stream>>>(dis, prob, gt, ws);
  k_final<<<1, 256, 0, stream>>>(prob, ws, out);
}